// FullConvRes_37434934952364
// MI455X (gfx1250) — compile-verified
//
#include <hip/hip_runtime.h>

// ---------------------------------------------------------------------------
// FullConvRes for MI455X (gfx1250, wave32, WMMA)
//   B=8, C=64, HW=4096, K=9
// Pipeline:
//   k0: rn[b,t]     = rsqrt(sum_c x[b,c,t]^2)                    (tiny)
//   k1: per (b, 16-row block): cos rows via V_WMMA_F32_16X16X4_F32,
//       kept in 256KB LDS; per-row bitonic argsort of u64 keys;
//       emit 9 sorted rank-indices -> sel[b,t,9]                 (dominant)
//   k2: gather 9 columns into LDS, 64x576x64 WMMA GEMM,
//       fused bias+relu+gamma*()+x residual -> out               (2.4 GFLOP)
// Workspace: rn (128KB) + sel (1.125MB) = ~1.3MB of d_ws.
// ---------------------------------------------------------------------------

typedef float v2f __attribute__((ext_vector_type(2)));
typedef float v8f __attribute__((ext_vector_type(8)));
typedef unsigned int u32;
typedef unsigned long long u64;

#define BN   8
#define CN   64
#define HWN  4096
#define KSEL 9
#define KK   (CN * KSEL)        // 576 reduction length for second GEMM
#define GSTR 72                 // padded LDS row stride: 2*72 % 64 == 16 -> no half-wave bank clash

// monotonic map: ascending uint order == ascending float order
__device__ __forceinline__ u32 fkey(float f) {
  u32 u = __float_as_uint(f);
  return (u & 0x80000000u) ? ~u : (u | 0x80000000u);
}

// ---------------------------------------------------------------- kernel 0 --
__global__ __launch_bounds__(256) void norm_kernel(const float* __restrict__ x,
                                                   float* __restrict__ rn) {
  int idx = blockIdx.x * 256 + threadIdx.x;   // idx = b*HWN + t
  int b = idx >> 12;
  int t = idx & (HWN - 1);
  const float* xb = x + (size_t)b * CN * HWN + t;
  float s = 0.f;
#pragma unroll 8
  for (int c = 0; c < CN; ++c) {
    float v = xb[(size_t)c * HWN];
    s += v * v;
  }
  rn[idx] = rsqrtf(s);
}

// ---------------------------------------------------------------- kernel 1 --
// grid = (HWN/16, B), block = 256 threads (8 waves)
// dyn LDS: cosk[16*4096] u32 (256KB) | skey[4096] u64 (32KB) | rnrow[16] | selrow[16]
__global__ __launch_bounds__(256) void cos_sort_select_kernel(
    const float* __restrict__ x, const float* __restrict__ rn,
    int* __restrict__ sel) {
  extern __shared__ char smem[];
  u32*   cosk   = (u32*)smem;                                   // 16*HWN u32
  u64*   skey   = (u64*)(smem + (size_t)16 * HWN * 4);          // HWN u64
  float* rnrow  = (float*)(smem + (size_t)16 * HWN * 4 + (size_t)HWN * 8);
  int*   selrow = (int*)(rnrow + 16);

  const int b    = blockIdx.y;
  const int t0   = blockIdx.x * 16;
  const int tid  = threadIdx.x;
  const int lane = tid & 31;
  const int wave = tid >> 5;
  const int hi   = (lane >> 4) & 1;   // which half of the wave
  const int lm   = lane & 15;

  const float* xb  = x + (size_t)b * CN * HWN;
  const float* rnb = rn + (size_t)b * HWN;

  if (tid < 16) rnrow[tid] = rnb[t0 + tid];
  __syncthreads();

  // ---- Preload A fragments for the whole row block (reused for all 256 col tiles)
  // A[m,k] = x[b, kb+k, t0+m];  lane holds M=lm, element j -> K = kb + 2*hi + j
  v2f afrag[16];
#pragma unroll
  for (int ks = 0; ks < 16; ++ks) {
    int kbase = ks * 4 + 2 * hi;
    afrag[ks][0] = xb[(size_t)kbase * HWN + t0 + lm];
    afrag[ks][1] = xb[(size_t)(kbase + 1) * HWN + t0 + lm];
  }

  // ---- GEMM: each wave sweeps column tiles s0 = wave*16 + 128*iter
  for (int s0 = wave * 16; s0 < HWN; s0 += 8 * 16) {
    v8f acc = {};
    float rs = rnb[s0 + lm];            // rn for column n = s0+lm (same for all v)
#pragma unroll
    for (int ks = 0; ks < 16; ++ks) {
      int kbase = ks * 4 + 2 * hi;
      v2f bfrag;
      bfrag[0] = xb[(size_t)kbase * HWN + s0 + lm];
      bfrag[1] = xb[(size_t)(kbase + 1) * HWN + s0 + lm];
      acc = __builtin_amdgcn_wmma_f32_16x16x4_f32(
          false, afrag[ks], false, bfrag, (short)0, acc, false, false);
    }
#pragma unroll
    for (int v = 0; v < 8; ++v) {
      int m = v + 8 * hi;               // D row
      float cv = acc[v] * rnrow[m] * rs;
      cosk[(size_t)m * HWN + s0 + lm] = fkey(cv);
    }
  }
  __syncthreads();

  // ---- per-row bitonic argsort on u64 keys (value<<32 | index == stable argsort)
  const int chunk = (HWN + KSEL - 1) / KSEL;   // 456
  for (int r = 0; r < 16; ++r) {
    for (int i = tid; i < HWN; i += 256)
      skey[i] = ((u64)cosk[(size_t)r * HWN + i] << 32) | (u32)i;
    __syncthreads();

    for (int k = 2; k <= HWN; k <<= 1) {
      for (int jj = k >> 1; jj > 0; jj >>= 1) {
        for (int i = tid; i < HWN; i += 256) {
          int ixj = i ^ jj;
          if (ixj > i) {
            u64 a = skey[i];
            u64 c = skey[ixj];
            bool up = ((i & k) == 0);
            if ((a > c) == up) { skey[i] = c; skey[ixj] = a; }
          }
        }
        __syncthreads();
      }
    }

    if (tid < KSEL) selrow[tid] = (int)(u32)(skey[(size_t)tid * chunk]);
    __syncthreads();
    if (tid == 0) {
      int v[KSEL];
#pragma unroll
      for (int q = 0; q < KSEL; ++q) v[q] = selrow[q];
      for (int q = 1; q < KSEL; ++q) {          // sel = sort(order[ranks])
        int key = v[q]; int p = q - 1;
        while (p >= 0 && v[p] > key) { v[p + 1] = v[p]; --p; }
        v[p + 1] = key;
      }
      int* so = sel + ((size_t)b * HWN + (t0 + r)) * KSEL;
#pragma unroll
      for (int q = 0; q < KSEL; ++q) so[q] = v[q];
    }
    __syncthreads();
  }
}

// ---------------------------------------------------------------- kernel 2 --
// grid = (HWN/64, B), block = 256 (8 waves); 16 output tiles of 16x16, 2/wave
// dyn LDS: G[KK][GSTR] floats (162KB) | selj[64*9] ints
__global__ __launch_bounds__(256) void conv_gather_kernel(
    const float* __restrict__ x, const float* __restrict__ w,
    const float* __restrict__ bias, const float* __restrict__ gamma,
    const int* __restrict__ sel, float* __restrict__ out) {
  extern __shared__ char smem[];
  float* G    = (float*)smem;                                 // KK x GSTR
  int*   selj = (int*)(smem + (size_t)KK * GSTR * 4);         // 64*KSEL

  const int b    = blockIdx.y;
  const int j0   = blockIdx.x * 64;
  const int tid  = threadIdx.x;
  const int lane = tid & 31;
  const int wave = tid >> 5;
  const int hi   = (lane >> 4) & 1;
  const int lm   = lane & 15;

  const float* xb   = x + (size_t)b * CN * HWN;
  const int*   selb = sel + ((size_t)b * HWN + j0) * KSEL;

  for (int i = tid; i < 64 * KSEL; i += 256) selj[i] = selb[i];
  __syncthreads();

  // gather G[kk][j] = x[b, c, sel[b, j0+j, k]] with kk = c*9+k (matches w flatten)
  for (int i = tid; i < KK * 64; i += 256) {
    int kk = i >> 6;
    int j  = i & 63;
    int c  = kk / KSEL;
    int k  = kk - c * KSEL;
    int s  = selj[j * KSEL + k];
    G[(size_t)kk * GSTR + j] = xb[(size_t)c * HWN + s];
  }
  __syncthreads();

  const float g = gamma[0];

  for (int tIdx = wave; tIdx < 16; tIdx += 8) {   // uniform: 2 tiles per wave
    int it  = tIdx >> 2;
    int jt  = tIdx & 3;
    int i0  = it * 16;
    int jj0 = jt * 16;
    v8f acc = {};
#pragma unroll 4
    for (int ks = 0; ks < KK / 4; ++ks) {         // 144 K-steps of 4
      int kb = ks * 4 + 2 * hi;
      v2f af, bf;
      af[0] = w[(size_t)(i0 + lm) * KK + kb];     // A[m,k] = weight[i0+m][kk]
      af[1] = w[(size_t)(i0 + lm) * KK + kb + 1];
      bf[0] = G[(size_t)kb * GSTR + jj0 + lm];    // B[k,n] from LDS
      bf[1] = G[(size_t)(kb + 1) * GSTR + jj0 + lm];
      acc = __builtin_amdgcn_wmma_f32_16x16x4_f32(
          false, af, false, bf, (short)0, acc, false, false);
    }
#pragma unroll
    for (int v = 0; v < 8; ++v) {
      int i = i0 + v + 8 * hi;
      int j = j0 + jj0 + lm;
      float cv = acc[v] + bias[i];
      cv = cv > 0.f ? cv : 0.f;
      size_t off = (size_t)b * CN * HWN + (size_t)i * HWN + j;
      out[off] = g * cv + x[off];
    }
  }
}

// ---------------------------------------------------------------- launcher --
extern "C" void kernel_launch(void* const* d_in, const int* in_sizes, int n_in,
                              void* d_out, int out_size, void* d_ws, size_t ws_size,
                              hipStream_t stream) {
  const float* x     = (const float*)d_in[0];   // (8,64,64,64)
  const float* w     = (const float*)d_in[1];   // (64,64,9)
  const float* bias  = (const float*)d_in[2];   // (64,)
  const float* gamma = (const float*)d_in[3];   // (1,)
  float* out = (float*)d_out;

  float* rn  = (float*)d_ws;                                    // B*HWN floats
  int*   sel = (int*)((char*)d_ws + (size_t)BN * HWN * sizeof(float)); // B*HWN*9 ints

  // k0: reciprocal norms
  norm_kernel<<<dim3((BN * HWN) / 256), 256, 0, stream>>>(x, rn);

  // k1: fused cosine GEMM (WMMA f32) + per-row bitonic argsort + rank select
  size_t smem1 = (size_t)16 * HWN * 4      // cos keys (256KB)
               + (size_t)HWN * 8           // sort keys (32KB)
               + 16 * 4 + 16 * 4;          // rnrow + selrow
  cos_sort_select_kernel<<<dim3(HWN / 16, BN), 256, smem1, stream>>>(x, rn, sel);

  // k2: gather + WMMA GEMM + epilogue
  size_t smem2 = (size_t)KK * GSTR * 4 + (size_t)64 * KSEL * 4;
  conv_gather_kernel<<<dim3(HWN / 64, BN), 256, smem2, stream>>>(x, w, bias, gamma, sel, out);
}